// WENONetwork_82240033783967
// MI455X (gfx1250) — compile-verified
//
#include <hip/hip_runtime.h>

// MI455X / gfx1250 (CDNA5, wave32). WENO5+WENO6 RHS with conv-net-modulated
// smoothness indicators.
//
// Roofline: ~50 MB traffic (2.2us @ 23.3 TB/s) vs ~4.2 GFLOP f32 -> compute
// bound on VALU; conv2 (3x12 GEMM per net) is offloaded to the f32 WMMA pipe
// (V_WMMA_F32_16X16X4_F32) which co-executes with the WENO VALU work.

typedef __attribute__((ext_vector_type(2))) float v2f;
typedef __attribute__((ext_vector_type(8))) float v8f;

#define TPB   256   // threads per block (8 wave32 waves)
#define PTS   256   // output points per block
#define YP    276   // activation row pitch (floats)
#define CNP   272   // conv-net output row pitch (floats)
#define TILES 17    // 17 x 16 = 272 conv-net positions cover the 258 needed

__device__ __forceinline__ float frcp(float x) { return __builtin_amdgcn_rcpf(x); }
__device__ __forceinline__ float fsigmoid(float x) { return frcp(1.f + __expf(-x)); }

__device__ __forceinline__ void omegas3(const float b[3], float d0, float d1, float d2,
                                        float e, float w[3]) {
  float brs = b[2] - b[0]; brs *= brs;
  float e0 = e + b[0], e1 = e + b[1], e2 = e + b[2];
  float q0 = e0 * e0, q1 = e1 * e1, q2 = e2 * e2;
  float w0 = d0 * frcp(q0) * (brs + q0);
  float w1 = d1 * frcp(q1) * (brs + q1);
  float w2 = d2 * frcp(q2) * (brs + q2);
  float inv = frcp(w0 + w1 + w2);
  w[0] = w0 * inv; w[1] = w1 * inv; w[2] = w2 * inv;
}

__device__ __forceinline__ void flux5(const float v[6], float f[3]) {
  const float i6 = 1.f / 6.f;
  f[0] = (11.f * v[3] - 7.f * v[4] + 2.f * v[5]) * i6;
  f[1] = (2.f * v[2] + 5.f * v[3] - v[4]) * i6;
  f[2] = (-v[1] + 5.f * v[2] + 2.f * v[3]) * i6;
}
__device__ __forceinline__ void beta5(const float v[6], float b[3]) {
  const float c = 13.f / 12.f;
  float t, s;
  t = v[3] - 2.f * v[4] + v[5]; s = 3.f * v[3] - 4.f * v[4] + v[5];
  b[0] = c * t * t + 0.25f * s * s;
  t = v[2] - 2.f * v[3] + v[4]; s = v[2] - v[4];
  b[1] = c * t * t + 0.25f * s * s;
  t = v[1] - 2.f * v[2] + v[3]; s = v[1] - 4.f * v[2] + 3.f * v[3];
  b[2] = c * t * t + 0.25f * s * s;
}
__device__ __forceinline__ void flux6(const float v[6], float f[3]) {
  const float i12 = 1.f / 12.f;
  f[0] = (v[0] - 3.f * v[1] - 9.f * v[2] + 11.f * v[3]) * i12;
  f[1] = (v[1] - 15.f * v[2] + 15.f * v[3] - v[4]) * i12;
  f[2] = (-11.f * v[2] + 9.f * v[3] + 3.f * v[4] - v[5]) * i12;
}
__device__ __forceinline__ void beta6(const float v[6], float b[3]) {
  const float c = 13.f / 12.f;
  float t, s;
  t = v[0] - 3.f * v[1] + 3.f * v[2] - v[3]; s = v[0] - 5.f * v[1] + 7.f * v[2] - 3.f * v[3];
  b[0] = c * t * t + 0.25f * s * s;
  t = v[1] - 3.f * v[2] + 3.f * v[3] - v[4]; s = v[1] - v[2] - v[3] + v[4];
  b[1] = c * t * t + 0.25f * s * s;
  t = v[2] - 3.f * v[3] + 3.f * v[4] - v[5]; s = -3.f * v[2] + 7.f * v[3] - 5.f * v[4] + v[5];
  b[2] = c * t * t + 0.25f * s * s;
}

__global__ __launch_bounds__(TPB) void weno_fused_kernel(
    const float* __restrict__ u,
    const float* __restrict__ w15, const float* __restrict__ b15,
    const float* __restrict__ w25, const float* __restrict__ b25,
    const float* __restrict__ w16, const float* __restrict__ b16,
    const float* __restrict__ w26, const float* __restrict__ b26,
    float* __restrict__ out, int N, int Mout)
{
  __shared__ float su[288];        // u[B0 .. B0+272]
  __shared__ float yL[8][YP];      // relu(conv1) activations, ch 0-3 net5, 4-7 net6
                                   // slot s = u-position (B0+1+s), s in [0,260)
  __shared__ float cnL[6][CNP];    // sigmoid(conv2): rows 0-2 net5, 3-5 net6
                                   // slot s = u-position (B0+2+s)

  const int tid  = threadIdx.x;
  const int lane = tid & 31;       // wave32
  const int wave = tid >> 5;       // 8 waves
  const int B0   = blockIdx.x * PTS;

  // ---- phase 1: stage u tile (each HBM byte read once per block) ----
  for (int t = tid; t < 273; t += TPB) {
    int g = B0 + t;
    su[t] = (g < N) ? u[g] : 0.f;
  }
  if (tid < 16) {                  // zero-fill activation slack read by tile 16
    #pragma unroll
    for (int c = 0; c < 8; ++c) yL[c][260 + tid] = 0.f;
  }
  __syncthreads();

  // ---- phase 2: conv1 (both nets) + ReLU, VALU (24 FMA/pos) ----
  {
    float W5[12], W6[12], Bb5[4], Bb6[4];
    #pragma unroll
    for (int k = 0; k < 12; ++k) { W5[k] = w15[k]; W6[k] = w16[k]; }
    #pragma unroll
    for (int k = 0; k < 4; ++k)  { Bb5[k] = b15[k]; Bb6[k] = b16[k]; }
    for (int p = tid; p < 260; p += TPB) {        // slot p = u-pos B0+1+p
      float x0 = su[p], x1 = su[p + 1], x2 = su[p + 2];
      #pragma unroll
      for (int c = 0; c < 4; ++c) {
        float a5 = Bb5[c] + W5[3*c]*x0 + W5[3*c+1]*x1 + W5[3*c+2]*x2;
        float a6 = Bb6[c] + W6[3*c]*x0 + W6[3*c+1]*x1 + W6[3*c+2]*x2;
        yL[c][p]     = fmaxf(a5, 0.f);
        yL[4 + c][p] = fmaxf(a6, 0.f);
      }
    }
  }
  __syncthreads();

  // ---- phase 3: conv2 + sigmoid via V_WMMA_F32_16X16X4_F32 ----
  // D = A x B per 16-position tile. A = weights (M rows: 0-2 net5 out,
  // 3-5 net6 out; K=24 block-diagonal over {ch,tap}, 6 chained K=4 steps).
  // B = shifted activation patches from LDS (N cols = positions).
  // f32 A layout (ISA 7.12.2): lane<16 row=lane, vgpr0=K0,vgpr1=K1;
  // lane>=16 same row, vgpr0=K2,vgpr1=K3.  B analogous: col=lane&15,
  // K = vgpr + 2*(lane>>4).  D: vgpr v = row v (lanes 0-15), col = lane.
  {
    const int row  = lane & 15;
    const int half = lane >> 4;
    v2f A[6];
    int bo[6][2];
    #pragma unroll
    for (int s = 0; s < 6; ++s) {
      #pragma unroll
      for (int c2 = 0; c2 < 2; ++c2) {
        int m = 4 * s + 2 * half + c2;            // K index 0..23
        float av = 0.f;
        if (m < 12) { if (row < 3) av = w25[row * 12 + m]; }
        else        { if (row >= 3 && row < 6) av = w26[(row - 3) * 12 + (m - 12)]; }
        if (c2 == 0) A[s].x = av; else A[s].y = av;
        int mm  = (m < 12) ? m : m - 12;
        int ch  = ((m < 12) ? 0 : 4) + mm / 3;
        int tap = mm - 3 * (mm / 3);
        bo[s][c2] = ch * YP + tap + row;          // + 16*t at use; col == row
      }
    }
    float bb[6] = { b25[0], b25[1], b25[2], b26[0], b26[1], b26[2] };
    const float* yF = &yL[0][0];
    for (int t = wave; t < TILES; t += 8) {       // wave-uniform: EXEC all-ones
      int base = 16 * t;
      v8f acc = {0.f, 0.f, 0.f, 0.f, 0.f, 0.f, 0.f, 0.f};
      #pragma unroll
      for (int s = 0; s < 6; ++s) {
        v2f bv;
        bv.x = yF[bo[s][0] + base];
        bv.y = yF[bo[s][1] + base];
        // (neg_a, A, neg_b, B, c_mod, C, reuse_a, reuse_b)
        acc = __builtin_amdgcn_wmma_f32_16x16x4_f32(
            false, A[s], false, bv, (short)0, acc, false, false);
      }
      if (lane < 16) {                            // rows 0..5 live in lanes 0..15
        int slot = base + lane;
        #pragma unroll
        for (int o = 0; o < 6; ++o)
          cnL[o][slot] = fsigmoid(acc[o] + bb[o]);
      }
    }
  }
  __syncthreads();

  // ---- phase 4: per-lane WENO5 + WENO6 algebra ----
  int i = B0 + tid;
  if (i < Mout) {
    const float e = 1e-13f;
    float lu[7];
    #pragma unroll
    for (int k = 0; k < 7; ++k) lu[k] = su[tid + k];
    float R[6], L[6];                             // R = ur stencil, L = ul stencil
    #pragma unroll
    for (int k = 0; k < 6; ++k) { R[k] = lu[k + 1]; L[k] = lu[k]; }

    float fR[3], fL[3], bR[3], bL[3], bp[3], bn[3];

    // WENO5 (conv at u-pos i+4 -> cn slot tid+2; i+2 -> slot tid)
    flux5(R, fR); flux5(L, fL); beta5(R, bR); beta5(L, bL);
    #pragma unroll
    for (int s = 0; s < 3; ++s) {
      bp[s] = bR[s] * cnL[s][tid + 2];
      bn[s] = bL[s] * cnL[s][tid];
    }
    float op[3], on[3];
    omegas3(bp, 0.1f, 0.6f, 0.3f, e, op);
    omegas3(bn, 0.1f, 0.6f, 0.3f, e, on);
    float rhs5 = op[0]*fR[0] + op[1]*fR[1] + op[2]*fR[2]
               - (on[0]*fL[0] + on[1]*fL[1] + on[2]*fL[2]);

    // WENO6
    flux6(R, fR); flux6(L, fL); beta6(R, bR); beta6(L, bL);
    #pragma unroll
    for (int s = 0; s < 3; ++s) {
      bp[s] = bR[s] * cnL[3 + s][tid + 2];
      bn[s] = bL[s] * cnL[3 + s][tid];
    }
    const float gp0 = 1.f/21.f, gp1 = 19.f/21.f, gp2 = 1.f/21.f;
    const float gn0 = 4.f/27.f, gn1 = 19.f/27.f, gn2 = 4.f/27.f;
    float opp[3], onp[3], opn[3], onn[3];
    omegas3(bp, gp0, gp1, gp2, e, opp);
    omegas3(bp, gn0, gn1, gn2, e, onp);
    omegas3(bn, gp0, gp1, gp2, e, opn);
    omegas3(bn, gn0, gn1, gn2, e, onn);
    const float sp = 42.f/15.f, sn = 27.f/15.f;
    float rhs6 = 0.f;
    #pragma unroll
    for (int s = 0; s < 3; ++s) {
      float wp = sp * opp[s] - sn * opn[s];
      float wn = sp * onp[s] - sn * onn[s];
      rhs6 += wp * fR[s] - wn * fL[s];
    }

    out[i] = rhs5;
    out[Mout + i] = rhs6;
  }
}

extern "C" void kernel_launch(void* const* d_in, const int* in_sizes, int n_in,
                              void* d_out, int out_size, void* d_ws, size_t ws_size,
                              hipStream_t stream) {
  const float* u   = (const float*)d_in[0];
  const float* w15 = (const float*)d_in[1];
  const float* b15 = (const float*)d_in[2];
  const float* w25 = (const float*)d_in[3];
  const float* b25 = (const float*)d_in[4];
  const float* w16 = (const float*)d_in[5];
  const float* b16 = (const float*)d_in[6];
  const float* w26 = (const float*)d_in[7];
  const float* b26 = (const float*)d_in[8];
  float* out = (float*)d_out;

  int N    = in_sizes[0];
  int Mout = out_size / 2;               // = N - 6
  int blocks = (Mout + PTS - 1) / PTS;

  weno_fused_kernel<<<blocks, TPB, 0, stream>>>(
      u, w15, b15, w25, b25, w16, b16, w26, b26, out, N, Mout);
}